// PRGNN_56049323212915
// MI455X (gfx1250) — compile-verified
//
#include <hip/hip_runtime.h>
#include <hip/hip_bf16.h>

typedef __attribute__((ext_vector_type(2))) float v2f;
typedef __attribute__((ext_vector_type(8))) float v8f;

#define FEAT 64

// ---------------- zero init (deg + pool accumulators) ----------------
__global__ void k_zero(float* __restrict__ deg, float* __restrict__ sums,
                       float* __restrict__ cnt, int n_nodes, int n_graphs) {
    int tid = blockIdx.x * blockDim.x + threadIdx.x;
    int total = n_nodes + 2 * n_graphs;
    if (tid >= total) return;
    if (tid < n_nodes) deg[tid] = 0.0f;
    else if (tid < n_nodes + n_graphs) sums[tid - n_nodes] = 0.0f;
    else cnt[tid - n_nodes - n_graphs] = 0.0f;
}

// ---------------- degree at dst ----------------
__global__ void k_degree(const int* __restrict__ dst, float* __restrict__ deg, int n_edges) {
    int e = blockIdx.x * blockDim.x + threadIdx.x;
    if (e >= n_edges) return;
    atomicAdd(&deg[dst[e]], 1.0f);
}

// ---------------- dinv = rsqrt(deg + 1) (self loop) ----------------
__global__ void k_dinv(float* __restrict__ deg, int n_nodes) {
    int i = blockIdx.x * blockDim.x + threadIdx.x;
    if (i >= n_nodes) return;
    deg[i] = rsqrtf(deg[i] + 1.0f);  // in place: deg buffer becomes dinv
}

// ---------------- W2^T so B fragments are contiguous float2 ----------------
__global__ void k_transpose64(const float* __restrict__ W, float* __restrict__ WT) {
    int tid = blockIdx.x * blockDim.x + threadIdx.x;
    if (tid >= FEAT * FEAT) return;
    int r = tid >> 6, c = tid & 63;
    WT[c * FEAT + r] = W[r * FEAT + c];
}

// ---------------- layer-1 GEMM (K=9, scalar) + dinv scale; writes hs and agg ----------------
__global__ void k_gemm1(const float* __restrict__ x, const float* __restrict__ W1,
                        const float* __restrict__ dinv, float* __restrict__ hs,
                        float* __restrict__ agg, int n_nodes) {
    int tid = blockIdx.x * blockDim.x + threadIdx.x;
    if (tid >= n_nodes * FEAT) return;
    int row = tid >> 6;
    int f = tid & 63;
    const float* xr = x + row * 9;
    float acc = 0.0f;
#pragma unroll
    for (int k = 0; k < 9; ++k) acc += xr[k] * W1[k * FEAT + f];
    float v = acc * dinv[row];
    hs[tid] = v;   // message source
    agg[tid] = v;  // self-loop term pre-seeded
}

// ---------------- edge scatter: agg[dst] += hs[src], float4 granularity ----------------
__global__ void k_scatter(const int* __restrict__ src, const int* __restrict__ dst,
                          const float* __restrict__ hs, float* __restrict__ agg,
                          unsigned int n_work) {
    unsigned int tid = blockIdx.x * blockDim.x + threadIdx.x;
    if (tid >= n_work) return;
    unsigned int e = tid >> 4;   // edge
    unsigned int c = tid & 15;   // float4 chunk within the 64-float row
    int s = src[e];
    int d = dst[e];
    const float4 v = *(const float4*)(hs + (size_t)s * FEAT + c * 4);
    float* p = agg + (size_t)d * FEAT + c * 4;
    atomicAdd(p + 0, v.x);
    atomicAdd(p + 1, v.y);
    atomicAdd(p + 2, v.z);
    atomicAdd(p + 3, v.w);
}

// ---------------- finalize layer: h = relu(dinv*agg + b) ----------------
__global__ void k_finalize(const float* __restrict__ agg, const float* __restrict__ dinv,
                           const float* __restrict__ b, float* __restrict__ h, int n_nodes) {
    int tid = blockIdx.x * blockDim.x + threadIdx.x;
    if (tid >= n_nodes * FEAT) return;
    int row = tid >> 6;
    int f = tid & 63;
    h[tid] = fmaxf(dinv[row] * agg[tid] + b[f], 0.0f);
}

// ---------------- layer-2 GEMM via V_WMMA_F32_16X16X4_F32 ----------------
// Block = 128 threads = 4 waves; block computes 32 rows x 64 cols.
// Each wave owns TWO 16x16 D tiles (M-split) sharing each B fragment, and
// iterates K=64 in 16 steps -> 32 WMMAs per wave, all loads are B64.
// A 16x4 layout: lanes 0-15 rows 0-15 hold K={0,1}; lanes 16-31 hold K={2,3}.
// B 4x16 layout mirrored (lane&15 = N; lane>>4 selects the K pair) — read
// from W2^T so the pair is a contiguous float2.
// D: VGPR r -> (M=r, N=lane) lanes 0-15, (M=r+8, N=lane-16) lanes 16-31.
__global__ __launch_bounds__(128) void k_gemm2_wmma(
    const float* __restrict__ h, const float* __restrict__ W2T,
    const float* __restrict__ dinv, float* __restrict__ hs,
    float* __restrict__ agg, int n_nodes) {
    const int lane = threadIdx.x & 31;
    const int wave = threadIdx.x >> 5;
    const int rowBase = blockIdx.x * 32;
    const int m = lane & 15;
    const int khalf = (lane >> 4) << 1;        // 0 or 2
    const int n = (wave << 4) + (lane & 15);   // output column 0..63
    const int nm1 = n_nodes - 1;
    // Clamp instead of predicating: out-of-range rows compute garbage that is
    // never stored. Keeps all loads unconditional (no exec-mask ladders).
    const int row0 = min(rowBase + m, nm1);
    const int row1 = min(rowBase + 16 + m, nm1);
    const float* __restrict__ a0p = h + (size_t)row0 * FEAT + khalf;
    const float* __restrict__ a1p = h + (size_t)row1 * FEAT + khalf;
    const float* __restrict__ bp  = W2T + (size_t)n * FEAT + khalf;

    v8f c0 = {0.f, 0.f, 0.f, 0.f, 0.f, 0.f, 0.f, 0.f};
    v8f c1 = {0.f, 0.f, 0.f, 0.f, 0.f, 0.f, 0.f, 0.f};
#pragma unroll
    for (int kk = 0; kk < 16; ++kk) {
        const v2f a0 = *(const v2f*)(a0p + (kk << 2));
        const v2f a1 = *(const v2f*)(a1p + (kk << 2));
        const v2f b  = *(const v2f*)(bp + (kk << 2));
        c0 = __builtin_amdgcn_wmma_f32_16x16x4_f32(
            false, a0, false, b, (short)0, c0, false, false);
        c1 = __builtin_amdgcn_wmma_f32_16x16x4_f32(
            false, a1, false, b, (short)0, c1, false, false);
    }

    const int rowHalf = (lane >> 4) << 3;  // +8 for upper-lane half
    const int dcol = (wave << 4) + (lane & 15);
#pragma unroll
    for (int r = 0; r < 8; ++r) {
        int dr = rowBase + r + rowHalf;
        if (dr < n_nodes) {
            float v = c0[r] * dinv[dr];
            hs[(size_t)dr * FEAT + dcol] = v;
            agg[(size_t)dr * FEAT + dcol] = v;
        }
        int dr2 = dr + 16;
        if (dr2 < n_nodes) {
            float v = c1[r] * dinv[dr2];
            hs[(size_t)dr2 * FEAT + dcol] = v;
            agg[(size_t)dr2 * FEAT + dcol] = v;
        }
    }
}

// ---------------- finalize2 + fc head + segment pool (one wave per node) ----------------
__global__ void k_head_pool(const float* __restrict__ agg, const float* __restrict__ dinv,
                            const float* __restrict__ b2, const float* __restrict__ Wfc,
                            const float* __restrict__ bfc, const int* __restrict__ batch,
                            float* __restrict__ sums, float* __restrict__ cnt, int n_nodes) {
    int lane = threadIdx.x & 31;
    int node = (blockIdx.x * blockDim.x + threadIdx.x) >> 5;
    if (node >= n_nodes) return;
    float di = dinv[node];
    float acc = 0.0f;
#pragma unroll
    for (int j = 0; j < 2; ++j) {
        int f = lane + 32 * j;
        float t = fmaxf(di * agg[(size_t)node * FEAT + f] + b2[f], 0.0f);
        acc += t * Wfc[f];
    }
#pragma unroll
    for (int off = 16; off > 0; off >>= 1) acc += __shfl_down(acc, off, 32);
    if (lane == 0) {
        int g = batch[node];
        atomicAdd(&sums[g], acc + bfc[0]);
        atomicAdd(&cnt[g], 1.0f);
    }
}

// ---------------- mean ----------------
__global__ void k_mean(const float* __restrict__ sums, const float* __restrict__ cnt,
                       float* __restrict__ out, int n_graphs) {
    int g = blockIdx.x * blockDim.x + threadIdx.x;
    if (g >= n_graphs) return;
    out[g] = sums[g] / fmaxf(cnt[g], 1.0f);
}

extern "C" void kernel_launch(void* const* d_in, const int* in_sizes, int n_in,
                              void* d_out, int out_size, void* d_ws, size_t ws_size,
                              hipStream_t stream) {
    const float* x     = (const float*)d_in[0];
    const int*   ei    = (const int*)d_in[1];
    const int*   batch = (const int*)d_in[2];
    const float* W1    = (const float*)d_in[4];
    const float* b1    = (const float*)d_in[5];
    const float* W2    = (const float*)d_in[6];
    const float* b2    = (const float*)d_in[7];
    const float* Wfc   = (const float*)d_in[8];
    const float* bfc   = (const float*)d_in[9];
    float* out = (float*)d_out;

    const int N = in_sizes[0] / 9;
    const int E = in_sizes[1] / 2;
    const int G = out_size;
    const int* src = ei;
    const int* dst = ei + E;

    // Workspace carve-up (all float, 16B-aligned pieces).
    float* ws = (float*)d_ws;
    float* dinv = ws;                         // N   (deg -> dinv in place)
    float* bufA = dinv + N;                   // N*64  hs1, later h2
    float* bufB = bufA + (size_t)N * FEAT;    // N*64  agg1, later agg2
    float* bufC = bufB + (size_t)N * FEAT;    // N*64  hs2
    float* sums = bufC + (size_t)N * FEAT;    // G
    float* cnt  = sums + G;                   // G
    float* W2T  = cnt + G;                    // 64*64

    const int B = 256;

    // 0) zero deg + pool accumulators (deterministic per call)
    {
        int n = N + 2 * G;
        k_zero<<<(n + B - 1) / B, B, 0, stream>>>(dinv, sums, cnt, N, G);
    }
    k_transpose64<<<(FEAT * FEAT + B - 1) / B, B, 0, stream>>>(W2, W2T);

    // 1) degree + dinv (shared by both conv layers)
    k_degree<<<(E + B - 1) / B, B, 0, stream>>>(dst, dinv, E);
    k_dinv<<<(N + B - 1) / B, B, 0, stream>>>(dinv, N);

    // 2) layer 1: hs1 = (x@W1)*dinv -> bufA, agg1 seeded with self-loop -> bufB
    k_gemm1<<<((size_t)N * FEAT + B - 1) / B, B, 0, stream>>>(x, W1, dinv, bufA, bufB, N);
    // 3) scatter: agg1[dst] += hs1[src]
    {
        unsigned int work = (unsigned int)E * 16u;
        k_scatter<<<(work + B - 1) / B, B, 0, stream>>>(src, dst, bufA, bufB, work);
    }
    // 4) h2 = relu(dinv*agg1 + b1) -> bufA (hs1 dead)
    k_finalize<<<((size_t)N * FEAT + B - 1) / B, B, 0, stream>>>(bufB, dinv, b1, bufA, N);

    // 5) layer 2 GEMM (WMMA f32): hs2 = (h2@W2)*dinv -> bufC, agg2 seeded -> bufB
    k_gemm2_wmma<<<(N + 31) / 32, 128, 0, stream>>>(bufA, W2T, dinv, bufC, bufB, N);
    // 6) scatter: agg2[dst] += hs2[src]
    {
        unsigned int work = (unsigned int)E * 16u;
        k_scatter<<<(work + B - 1) / B, B, 0, stream>>>(src, dst, bufC, bufB, work);
    }
    // 7) head + pool: per-node relu/fc, segment sums
    k_head_pool<<<((size_t)N * 32 + B - 1) / B, B, 0, stream>>>(bufB, dinv, b2, Wfc, bfc,
                                                               batch, sums, cnt, N);
    // 8) mean
    k_mean<<<(G + B - 1) / B, B, 0, stream>>>(sums, cnt, out, G);
}